// GCNGraphSAGENodeSetUpdate_14199161880653
// MI455X (gfx1250) — compile-verified
//
#include <hip/hip_runtime.h>
#include <hip/hip_bf16.h>

typedef __attribute__((ext_vector_type(2))) float v2f;
typedef __attribute__((ext_vector_type(8))) float v8f;

#define N_PAPER 100000
#define DIM 128
#define UNITS 128
#define LDS_STRIDE 132   // 128 + 4 pad floats: row-stride bank offset = 132 % 64 = 4 -> no conflicts
#define MT 5             // M-tiles per block (80 rows); 100000 / 80 = 1250 exact
#define ROWS (MT * 16)

// ---------------------------------------------------------------------------
// Edge scatter: agg[dst] += x[src], deg[dst] += 1.  One wave (32 lanes) per
// edge; each lane moves float4 (16B) and does 4 f32 global atomics (these
// land in the 192MB L2 -- the 51MB destination array is L2-resident).
// ---------------------------------------------------------------------------
__global__ void __launch_bounds__(256)
scatter_edges(const float* __restrict__ x,
              const int* __restrict__ src,
              const int* __restrict__ dst,
              float* __restrict__ agg,
              float* __restrict__ deg,
              int n_edges)
{
    int edge = blockIdx.x * 8 + (threadIdx.x >> 5);
    if (edge >= n_edges) return;
    int lane = threadIdx.x & 31;

    int s = src[edge];
    int d = dst[edge];

    const float4 v = *(const float4*)(x + (size_t)s * DIM + lane * 4);
    float* o = agg + (size_t)d * DIM + lane * 4;
    unsafeAtomicAdd(o + 0, v.x);
    unsafeAtomicAdd(o + 1, v.y);
    unsafeAtomicAdd(o + 2, v.z);
    unsafeAtomicAdd(o + 3, v.w);
    if (lane == 0) unsafeAtomicAdd(deg + d, 1.0f);
}

// ---------------------------------------------------------------------------
// Fused triple-GEMM + mean + bias + ReLU epilogue.
//   out = relu((agg_c@W_c + agg_w@W_w + x_paper@W_s) / (deg+1) + bias)
// Block = 256 threads = 8 waves, owns an 80-row stripe (5 M-tiles).  Each
// wave owns one 16-wide N-tile and 5 accumulators; each B fragment (2 scalar
// loads of W) is reused by 5 v_wmma_f32_16x16x4_f32, and the 5 independent
// accumulation chains interleave to cover XDL latency.  The three A matrices
// are streamed sequentially through one 42KB LDS tile.
// ---------------------------------------------------------------------------
__global__ void __launch_bounds__(256)
gemm_mean_relu(const float* __restrict__ x_paper,
               const float* __restrict__ agg_c,
               const float* __restrict__ agg_w,
               const float* __restrict__ W_cites,
               const float* __restrict__ W_writes,
               const float* __restrict__ W_self,
               const float* __restrict__ bias,
               const float* __restrict__ deg,
               float* __restrict__ out)
{
    __shared__ float lds[ROWS * LDS_STRIDE];   // 80*132*4 = 42240 B

    const int base_row = blockIdx.x * ROWS;
    const int tid  = threadIdx.x;
    const int wave = tid >> 5;          // N-tile index 0..7
    const int lane = tid & 31;
    const int half = lane >> 4;         // lane half selects K pair / M half
    const int mrow = lane & 15;         // A fragment: lane = M row
    const int col  = wave * 16 + mrow;  // B/C fragment: lane = N col

    v8f acc[MT] = {};

    const float* Amats[3] = { agg_c,   agg_w,    x_paper };
    const float* Wmats[3] = { W_cites, W_writes, W_self  };

    for (int m = 0; m < 3; ++m) {
        if (m) __syncthreads();         // previous compute done before overwrite

        // ---- stage 80x128 A-tile into LDS (float4, coalesced) ----
        const float* A = Amats[m];
        #pragma unroll
        for (int s = tid; s < ROWS * 32; s += 256) {   // 80 rows * 32 float4
            int row  = s >> 5;
            int colv = (s & 31) * 4;
            float4 v = *(const float4*)(A + (size_t)(base_row + row) * DIM + colv);
            *(float4*)(&lds[row * LDS_STRIDE + colv]) = v;
        }
        __syncthreads();

        // ---- K loop: one B fragment feeds 5 wmma ----
        const float* W = Wmats[m];
        #pragma unroll
        for (int k0 = 0; k0 < DIM; k0 += 4) {
            int k = k0 + 2 * half;
            v2f b;                                     // B 4x16 fragment
            b.x = W[(size_t)(k + 0) * UNITS + col];
            b.y = W[(size_t)(k + 1) * UNITS + col];
            #pragma unroll
            for (int t = 0; t < MT; ++t) {
                // A 16x4 fragment: VGPR0 = A[m][k], VGPR1 = A[m][k+1]
                v2f a = *(const v2f*)(&lds[(t * 16 + mrow) * LDS_STRIDE + k]);
                acc[t] = __builtin_amdgcn_wmma_f32_16x16x4_f32(
                    false, a, false, b, (short)0, acc[t], false, false);
            }
        }
    }

    // ---- epilogue: mean over (deg+1), + bias, ReLU ----
    const float bcol = bias[col];
    #pragma unroll
    for (int t = 0; t < MT; ++t) {
        #pragma unroll
        for (int i = 0; i < 8; ++i) {
            int row = base_row + t * 16 + i + 8 * half;  // C layout: VGPRi -> M=i / M=i+8
            float dnv = deg[row] + 1.0f;                 // self-loop counts as one edge
            float val = acc[t][i] / dnv + bcol;
            out[(size_t)row * UNITS + col] = fmaxf(val, 0.0f);
        }
    }
}

// ---------------------------------------------------------------------------
extern "C" void kernel_launch(void* const* d_in, const int* in_sizes, int n_in,
                              void* d_out, int out_size, void* d_ws, size_t ws_size,
                              hipStream_t stream)
{
    const float* x_paper    = (const float*)d_in[0];
    const float* x_author   = (const float*)d_in[1];
    const float* W_cites    = (const float*)d_in[2];
    const float* W_writes   = (const float*)d_in[3];
    const float* W_self     = (const float*)d_in[4];
    const float* bias       = (const float*)d_in[5];
    const int*   cites_src  = (const int*)d_in[6];
    const int*   cites_dst  = (const int*)d_in[7];
    const int*   writes_src = (const int*)d_in[8];
    const int*   writes_dst = (const int*)d_in[9];
    float* out = (float*)d_out;

    // Workspace layout: agg_c [100000*128] | agg_w [100000*128] | deg [100000]
    float* agg_c = (float*)d_ws;
    float* agg_w = agg_c + (size_t)N_PAPER * DIM;
    float* degp  = agg_w + (size_t)N_PAPER * DIM;
    size_t zero_bytes = ((size_t)N_PAPER * DIM * 2 + N_PAPER) * sizeof(float);
    hipMemsetAsync(d_ws, 0, zero_bytes, stream);

    const int E = in_sizes[6];          // 500000 edges per set
    dim3 blk(256);

    // segment_sum is linear: scatter raw features, GEMM once afterwards.
    scatter_edges<<<(E + 7) / 8, blk, 0, stream>>>(x_paper,  cites_src,  cites_dst,  agg_c, degp, E);
    scatter_edges<<<(E + 7) / 8, blk, 0, stream>>>(x_author, writes_src, writes_dst, agg_w, degp, E);

    gemm_mean_relu<<<N_PAPER / ROWS, blk, 0, stream>>>(
        x_paper, agg_c, agg_w, W_cites, W_writes, W_self, bias, degp, out);
}